// FillNet_1288490189631
// MI455X (gfx1250) — compile-verified
//
#include <hip/hip_runtime.h>

typedef __attribute__((ext_vector_type(2))) float v2f;
typedef __attribute__((ext_vector_type(8))) float v8f;

#define BLOCK_THREADS 256
#define WAVES_PER_BLOCK 8   // wave32: 256 threads = 8 waves

// K(x,p) = exp(-d^2 / (2*sigma^2)), sigma = 1.2  ->  exp2(d^2 * SCALE)
// SCALE = -log2(e) / (2*1.2^2) = -1.4426950408889634 / 2.88
#define RBF_SCALE (-0.50093577808644563f)

// One wave handles a 16-row tile of X against all P patterns.
// Per chunk of 4 patterns:
//   B (4x16 f32)  : kernel values K[pattern k][x-point n]
//       lane L (0-15): v0 = K[p+0][L],  v1 = K[p+1][L]
//       lane L (16-31): v0 = K[p+2][L-16], v1 = K[p+3][L-16]
//   A (16x4 f32)  : row M=0 = W2[p..p+3], row M=1 = ones, rows 2..15 = 0
//       lane 0: (W2[p],W2[p+1])  lane 16: (W2[p+2],W2[p+3])
//       lanes 1,17: (1,1)        else: (0,0)
//   D = A*B + C : D[0][n] accumulates num, D[1][n] accumulates den.
// Final: lanes 0-15 hold num in acc[0] (M=0) and den in acc[1] (M=1).
__global__ __launch_bounds__(BLOCK_THREADS)
void rbf_reduce_wmma_kernel(const float* __restrict__ X,
                            const float* __restrict__ pat,
                            const float* __restrict__ W2,
                            float* __restrict__ out,
                            int N, int P)
{
    extern __shared__ float smem[];
    float* s_pat = smem;            // P*2 floats (64 KB for P=8192)
    float* s_w   = smem + 2 * P;    // P floats   (32 KB)

    const int tid = threadIdx.x;

    // Stage patterns + weights into LDS (float4 vector copies, no divergence).
    {
        const float4* gp4 = (const float4*)pat;
        float4*       sp4 = (float4*)s_pat;
        const int np4 = (P * 2) >> 2;
        for (int i = tid; i < np4; i += BLOCK_THREADS) sp4[i] = gp4[i];

        const float4* gw4 = (const float4*)W2;
        float4*       sw4 = (float4*)s_w;
        const int nw4 = P >> 2;
        for (int i = tid; i < nw4; i += BLOCK_THREADS) sw4[i] = gw4[i];
    }
    __syncthreads();

    const int lane = tid & 31;
    const int wave = tid >> 5;
    const int tile = blockIdx.x * WAVES_PER_BLOCK + wave;   // 16 X rows
    const int msub = lane & 15;          // x-point column / A-matrix row id
    const int hi   = lane >> 4;          // 0 -> K=0,1 ; 1 -> K=2,3

    // This lane's X point (B-matrix column N = msub). Lanes 16-31 duplicate.
    const float2 xv = ((const float2*)X)[tile * 16 + msub];
    const float x0 = xv.x, x1 = xv.y;

    // Loop-invariant A-row selection for this lane.
    const bool  laneIsW = (msub == 0);                   // weights row
    const float aFill   = (msub == 1) ? 1.0f : 0.0f;     // ones row / zeros

    const float4* sp4 = (const float4*)s_pat;   // 2 patterns per float4
    const float2* sw2 = (const float2*)s_w;     // 2 weights per float2

    v8f acc0 = {};
    v8f acc1 = {};

    #pragma unroll 4
    for (int p = 0; p < P; p += 8) {
        // chunk 0: patterns p .. p+3 -> acc0
        {
            const int j = (p >> 1) + hi;
            const float4 q = sp4[j];            // patterns (q.x,q.y),(q.z,q.w)
            const float2 w = sw2[j];
            const float dx0 = x0 - q.x, dy0 = x1 - q.y;
            const float dx1 = x0 - q.z, dy1 = x1 - q.w;
            const float d0 = dx0 * dx0 + dy0 * dy0;
            const float d1 = dx1 * dx1 + dy1 * dy1;
            v2f b = { __builtin_amdgcn_exp2f(d0 * RBF_SCALE),
                      __builtin_amdgcn_exp2f(d1 * RBF_SCALE) };
            v2f a = { laneIsW ? w.x : aFill,
                      laneIsW ? w.y : aFill };
            acc0 = __builtin_amdgcn_wmma_f32_16x16x4_f32(
                       false, a, false, b, (short)0, acc0, false, false);
        }
        // chunk 1: patterns p+4 .. p+7 -> acc1 (independent accumulator)
        {
            const int j = (p >> 1) + 2 + hi;
            const float4 q = sp4[j];
            const float2 w = sw2[j];
            const float dx0 = x0 - q.x, dy0 = x1 - q.y;
            const float dx1 = x0 - q.z, dy1 = x1 - q.w;
            const float d0 = dx0 * dx0 + dy0 * dy0;
            const float d1 = dx1 * dx1 + dy1 * dy1;
            v2f b = { __builtin_amdgcn_exp2f(d0 * RBF_SCALE),
                      __builtin_amdgcn_exp2f(d1 * RBF_SCALE) };
            v2f a = { laneIsW ? w.x : aFill,
                      laneIsW ? w.y : aFill };
            acc1 = __builtin_amdgcn_wmma_f32_16x16x4_f32(
                       false, a, false, b, (short)0, acc1, false, false);
        }
    }

    // D layout: lanes 0-15 VGPR i <-> row M=i.  num = M0, den = M1.
    const float num = acc0[0] + acc1[0];
    const float den = acc0[1] + acc1[1];
    if (lane < 16) {
        out[tile * 16 + lane] = num / den;
    }
}

extern "C" void kernel_launch(void* const* d_in, const int* in_sizes, int n_in,
                              void* d_out, int out_size, void* d_ws, size_t ws_size,
                              hipStream_t stream) {
    (void)n_in; (void)out_size; (void)d_ws; (void)ws_size;

    const float* X   = (const float*)d_in[0];   // [N,2]
    const float* pat = (const float*)d_in[1];   // [P,2]
    const float* W2  = (const float*)d_in[2];   // [P]
    float*       out = (float*)d_out;           // [N]

    const int N = in_sizes[0] / 2;              // 32768
    const int P = in_sizes[2];                  // 8192

    const int tiles  = N / 16;                  // 2048 wave-tiles
    const int blocks = tiles / WAVES_PER_BLOCK; // 256 blocks of 8 waves
    const size_t lds_bytes = (size_t)P * 3 * sizeof(float); // 96 KB

    rbf_reduce_wmma_kernel<<<blocks, BLOCK_THREADS, lds_bytes, stream>>>(
        X, pat, W2, out, N, P);
}